// LSTM_39934605918264
// MI455X (gfx1250) — compile-verified
//
#include <hip/hip_runtime.h>
#include <hip/hip_bf16.h>
#include <math.h>

typedef __attribute__((ext_vector_type(2))) float v2f;
typedef __attribute__((ext_vector_type(8))) float v8f;

#define B_   256
#define T_   512
#define E_   200
#define FX   203      // x feature count (200 + 3)
#define KPAD 204      // pad K to multiple of 4
#define H_   24
#define G4   96       // 4*H

__device__ __forceinline__ float sigf(float x) {
    return 1.0f / (1.0f + __expf(-x));
}

// ---------------------------------------------------------------------------
// Kernel A: Xpre[t][n][b] = ([Ew[word], Ec[cap]] @ W_lstm[0:203] + bias)[b][n]
// Layout is [T][96][B] so that one lane's 8 accumulator elements (rows
// b0+v+8*hi) are contiguous -> float4 stores here, float4 loads in kernel B.
// One wave per 16-row tile; f32 WMMA 16x16x4, K looped over 204.
// ---------------------------------------------------------------------------
__global__ __launch_bounds__(32) void lstm_xproj_kernel(
    const int* __restrict__ words, const int* __restrict__ caps,
    const float* __restrict__ Ew, const float* __restrict__ Ec,
    const float* __restrict__ W, const float* __restrict__ bias,
    float* __restrict__ Xpre)
{
    __shared__ float As[16][KPAD];   // 16 rows x 204 gathered features (816B rows, 16B aligned)
    __shared__ float Wt[KPAD][16];   // 204 x 16 weight col-tile (64B rows)

    const int lane = threadIdx.x;           // 0..31
    const int rowbase = blockIdx.x * 16;    // rows r = t*B + b (B%16==0 -> same t)
    const int trow  = rowbase / B_;
    const int bbase = rowbase % B_;

    // Stage A tile: 2 lanes per row, float4 loads (Ew rows are 800B, 16B aligned).
    {
        const int m    = lane >> 1;
        const int half = lane & 1;
        const int b = bbase + m;
        const int w  = words[b * T_ + trow];
        const int cp = caps [b * T_ + trow];
        const float4* ewrow = (const float4*)(Ew + (size_t)w * E_);
        // 50 float4 per row; lane half takes alternating quads (25 each).
        for (int j = 0; j < 25; ++j) {
            int q = j * 2 + half;            // 0..49
            *(float4*)&As[m][q * 4] = ewrow[q];
        }
        if (half == 0) {                      // capitals features + K padding
            float4 tail;
            tail.x = Ec[cp * 3 + 0];
            tail.y = Ec[cp * 3 + 1];
            tail.z = Ec[cp * 3 + 2];
            tail.w = 0.0f;
            *(float4*)&As[m][E_] = tail;      // cols 200..203
        }
    }
    __syncthreads();

    const int mrow = lane & 15;
    const int hi   = lane >> 4;             // 0/1: selects K pair / M half

    for (int tl = 0; tl < 6; ++tl) {
        // Stage this 16-column slice of W (float4; rows 0..202, row 203 zeroed).
        for (int idx = lane; idx < KPAD * 4; idx += 32) {
            int k = idx >> 2, q = idx & 3;
            float4 v = {0.0f, 0.0f, 0.0f, 0.0f};
            if (k < FX) v = *(const float4*)(W + (size_t)k * G4 + tl * 16 + q * 4);
            *(float4*)&Wt[k][q * 4] = v;
        }
        __syncthreads();

        const float bv = bias[tl * 16 + mrow];
        v8f acc = {bv, bv, bv, bv, bv, bv, bv, bv};   // bias folded into C

        for (int kb = 0; kb < KPAD; kb += 4) {
            v2f a, bm;
            a.x  = As[mrow][kb + 2 * hi + 0];
            a.y  = As[mrow][kb + 2 * hi + 1];
            bm.x = Wt[kb + 2 * hi + 0][mrow];
            bm.y = Wt[kb + 2 * hi + 1][mrow];
            acc = __builtin_amdgcn_wmma_f32_16x16x4_f32(
                false, a, false, bm, (short)0, acc, false, false);
        }

        // Store: Xpre[(t*96 + n)*B + b], b = bbase + v + 8*hi contiguous in v.
        {
            size_t base = ((size_t)trow * G4 + tl * 16 + mrow) * B_ + bbase + 8 * hi;
            float4 lo = {acc[0], acc[1], acc[2], acc[3]};
            float4 hh = {acc[4], acc[5], acc[6], acc[7]};
            *(float4*)&Xpre[base + 0] = lo;
            *(float4*)&Xpre[base + 4] = hh;
        }
        __syncthreads();   // protect Wt before next tile restage
    }
}

// ---------------------------------------------------------------------------
// Kernel B: sequential recurrence. One wave per 16-batch-row tile (16 blocks).
// z = Xpre[t] + h @ W_h (WMMA f32, K=24), gates in TF order i,j,f,o.
// Next step's Xpre tile is prefetched (global_prefetch) to hide latency.
// ---------------------------------------------------------------------------
__global__ __launch_bounds__(32) void lstm_recurrence_kernel(
    const float* __restrict__ W,       // full W_lstm [227][96]
    const float* __restrict__ Xpre,    // [T][96][B]
    float* __restrict__ hmean)         // [B][24]
{
    __shared__ float Whs[H_][100];     // 24 x 100 (400B rows -> b128 staging OK)
    __shared__ float zs[16][G4 + 1];   // 16 x 97 gate pre-activations
    __shared__ float hs[16][H_ + 2];   // 16 x 26 hidden state

    const int lane = threadIdx.x;
    const int b0   = blockIdx.x * 16;

    // Stage recurrent weights (rows 203..226) with float4.
    for (int idx = lane; idx < H_ * (G4 / 4); idx += 32) {
        int k = idx / (G4 / 4), q = idx % (G4 / 4);
        *(float4*)&Whs[k][q * 4] =
            *(const float4*)(W + (size_t)(FX + k) * G4 + q * 4);
    }
    for (int idx = lane; idx < 16 * (H_ + 2); idx += 32)
        (&hs[0][0])[idx] = 0.0f;
    __syncthreads();

    const int mrow = lane & 15;
    const int hi   = lane >> 4;
    // Per-lane gate ownership: row gm = lane/2, 12 columns starting gn0.
    const int gm  = lane >> 1;
    const int gn0 = (lane & 1) * 12;
    float creg[12], hsum[12];
    for (int j = 0; j < 12; ++j) { creg[j] = 0.0f; hsum[j] = 0.0f; }

    for (int t = 0; t < T_; ++t) {
        // A-fragments of h (16x24) from LDS.
        v2f afr[6];
        for (int kk = 0; kk < 6; ++kk) {
            afr[kk].x = hs[mrow][kk * 4 + 2 * hi + 0];
            afr[kk].y = hs[mrow][kk * 4 + 2 * hi + 1];
        }
        // Seed accumulators with x-projection (incl. bias): float4 pairs.
        const float* xt = Xpre + (size_t)t * G4 * B_;
        v8f acc[6];
        for (int tl = 0; tl < 6; ++tl) {
            size_t base = (size_t)(tl * 16 + mrow) * B_ + b0 + 8 * hi;
            float4 lo = *(const float4*)(xt + base + 0);
            float4 hh = *(const float4*)(xt + base + 4);
            acc[tl][0] = lo.x; acc[tl][1] = lo.y; acc[tl][2] = lo.z; acc[tl][3] = lo.w;
            acc[tl][4] = hh.x; acc[tl][5] = hh.y; acc[tl][6] = hh.z; acc[tl][7] = hh.w;
        }
        // Prefetch next step's tile (96 chunks of 64B at stride 1KB).
        {
            int tn = (t + 1 < T_) ? (t + 1) : t;
            const float* xn = Xpre + (size_t)tn * G4 * B_ + b0;
            __builtin_prefetch(xn + (size_t)(lane * 3 + 0) * B_, 0, 3);
            __builtin_prefetch(xn + (size_t)(lane * 3 + 1) * B_, 0, 3);
            __builtin_prefetch(xn + (size_t)(lane * 3 + 2) * B_, 0, 3);
        }

        for (int kk = 0; kk < 6; ++kk) {
            for (int tl = 0; tl < 6; ++tl) {
                v2f bm;
                bm.x = Whs[kk * 4 + 2 * hi + 0][tl * 16 + mrow];
                bm.y = Whs[kk * 4 + 2 * hi + 1][tl * 16 + mrow];
                acc[tl] = __builtin_amdgcn_wmma_f32_16x16x4_f32(
                    false, afr[kk], false, bm, (short)0, acc[tl], false, false);
            }
        }
        // Publish z tiles to LDS for gate-wise reshuffle.
        for (int tl = 0; tl < 6; ++tl)
            for (int v = 0; v < 8; ++v)
                zs[v + 8 * hi][tl * 16 + mrow] = acc[tl][v];
        __syncthreads();

        // Gate math: c = sig(f+1)*c + sig(i)*tanh(j); h = sig(o)*tanh(c)
        for (int j = 0; j < 12; ++j) {
            int n = gn0 + j;
            float iz = zs[gm][n];
            float jz = zs[gm][n + 24];
            float fz = zs[gm][n + 48];
            float oz = zs[gm][n + 72];
            float c = sigf(fz + 1.0f) * creg[j] + sigf(iz) * tanhf(jz);
            float h = sigf(oz) * tanhf(c);
            creg[j] = c;
            hsum[j] += h;
            hs[gm][n] = h;
        }
        __syncthreads();   // h visible before next step's A-fragment read
    }

    for (int j = 0; j < 12; ++j)
        hmean[(size_t)(b0 + gm) * H_ + gn0 + j] = hsum[j] * (1.0f / T_);
}

// ---------------------------------------------------------------------------
// Kernel C: head. out = sigmoid(elu(hmean @ W1 + b1) @ W2 + b2). 1 row/thread.
// ---------------------------------------------------------------------------
__global__ __launch_bounds__(256) void lstm_head_kernel(
    const float* __restrict__ hmean,
    const float* __restrict__ W1, const float* __restrict__ b1,
    const float* __restrict__ W2, const float* __restrict__ b2,
    float* __restrict__ out)
{
    int b = blockIdx.x * blockDim.x + threadIdx.x;
    if (b >= B_) return;
    float hv[H_];
    for (int k = 0; k < H_; ++k) hv[k] = hmean[b * H_ + k];
    float l1[50];
    for (int j = 0; j < 50; ++j) {
        float s = b1[j];
        for (int k = 0; k < H_; ++k) s += hv[k] * W1[k * 50 + j];
        l1[j] = (s > 0.0f) ? s : (__expf(s) - 1.0f);   // elu
    }
    for (int j = 0; j < 6; ++j) {
        float s = b2[j];
        for (int k = 0; k < 50; ++k) s += l1[k] * W2[k * 6 + j];
        out[b * 6 + j] = sigf(s);
    }
}

extern "C" void kernel_launch(void* const* d_in, const int* in_sizes, int n_in,
                              void* d_out, int out_size, void* d_ws, size_t ws_size,
                              hipStream_t stream) {
    const int*   words = (const int*)  d_in[0];
    const int*   caps  = (const int*)  d_in[1];
    const float* Ew    = (const float*)d_in[2];
    const float* Ec    = (const float*)d_in[3];
    const float* Wl    = (const float*)d_in[4];
    const float* bl    = (const float*)d_in[5];
    const float* W1    = (const float*)d_in[6];
    const float* b1    = (const float*)d_in[7];
    const float* W2    = (const float*)d_in[8];
    const float* b2    = (const float*)d_in[9];
    float* out = (float*)d_out;

    // Workspace layout: Xpre [T][96][B] then hmean [B][24]
    float* Xpre  = (float*)d_ws;
    float* hmean = Xpre + (size_t)T_ * G4 * B_;

    // A: parallel x-projection GEMM: (B*T)/16 = 8192 row tiles, 1 wave each.
    lstm_xproj_kernel<<<(B_ * T_) / 16, 32, 0, stream>>>(
        words, caps, Ew, Ec, Wl, bl, Xpre);

    // B: sequential recurrence, one wave per 16 batch rows.
    lstm_recurrence_kernel<<<B_ / 16, 32, 0, stream>>>(Wl, Xpre, hmean);

    // C: classifier head.
    lstm_head_kernel<<<1, 256, 0, stream>>>(hmean, W1, b1, W2, b2, out);
}